// GATLayer_80281528697219
// MI455X (gfx1250) — compile-verified
//
#include <hip/hip_runtime.h>
#include <math.h>

typedef __attribute__((ext_vector_type(2))) float v2f;
typedef __attribute__((ext_vector_type(8))) float v8f;

#define HIDDEN 64
#define RES_F 0.5f

// ---------------------------------------------------------------------------
// Kernel 1: per-node transform using V_WMMA_F32_16X16X4_F32.
// One wave handles a 16-node tile: computes relu(X@W + b) for W1 and W2
// (16x64 output tiles via 4 N-blocks x 16 K-steps of 16x16x4 WMMA),
// immediately contracts with att_w to the per-node scalars s1, s2.
// ---------------------------------------------------------------------------
__global__ void __launch_bounds__(128)
node_wmma_kernel(const float* __restrict__ embeds,
                 const float* __restrict__ W1, const float* __restrict__ b1,
                 const float* __restrict__ W2, const float* __restrict__ b2,
                 const float* __restrict__ attw,
                 float* __restrict__ s1, float* __restrict__ s2,
                 int n_nodes, int n_tiles)
{
    const int wave = blockIdx.x * (blockDim.x >> 5) + (threadIdx.x >> 5);
    if (wave >= n_tiles) return;                 // wave-uniform: EXEC stays all-1s
    const int lane = threadIdx.x & 31;
    const int half = lane >> 4;                  // 0: K={0,1}; 1: K={2,3} (per 16x4 f32 A layout)
    const int l16  = lane & 15;

    // A fragment: row M = l16 of the tile, columns k*4 + 2*half + {0,1}
    int arow = wave * 16 + l16;
    if (arow >= n_nodes) arow = n_nodes - 1;     // clamp (N=50000 is a multiple of 16 anyway)
    const float* Arow = embeds + (size_t)arow * HIDDEN + 2 * half;
    v2f afrag[16];
#pragma unroll
    for (int k = 0; k < 16; ++k) {
        afrag[k].x = Arow[k * 4 + 0];
        afrag[k].y = Arow[k * 4 + 1];
    }

#pragma unroll
    for (int pass = 0; pass < 2; ++pass) {
        const float* __restrict__ W    = pass ? W2 : W1;
        const float* __restrict__ bias = pass ? b2 : b1;
        const float* __restrict__ aw   = attw + pass * HIDDEN;
        float* __restrict__ sout       = pass ? s2 : s1;

        float p[8];
#pragma unroll
        for (int r = 0; r < 8; ++r) p[r] = 0.0f;

        for (int nb = 0; nb < 4; ++nb) {
            const int n = nb * 16 + l16;         // output column (same for both halves)
            v8f acc = {0.f, 0.f, 0.f, 0.f, 0.f, 0.f, 0.f, 0.f};
#pragma unroll
            for (int k = 0; k < 16; ++k) {
                // B fragment: 4x16 block, rows k*4 + 2*half + {0,1}, column n
                v2f bfrag;
                bfrag.x = W[(size_t)(k * 4 + 2 * half + 0) * HIDDEN + n];
                bfrag.y = W[(size_t)(k * 4 + 2 * half + 1) * HIDDEN + n];
                acc = __builtin_amdgcn_wmma_f32_16x16x4_f32(
                    /*neg_a=*/false, afrag[k],
                    /*neg_b=*/false, bfrag,
                    /*c_mod=*/(short)0, acc,
                    /*reuse_a=*/false, /*reuse_b=*/false);
            }
            // D layout: VGPR r -> M = r + 8*half, N = n. Fuse bias + relu + att_w dot.
            const float bn = bias[n];
            const float an = aw[n];
#pragma unroll
            for (int r = 0; r < 8; ++r) {
                float v = acc[r] + bn;
                v = v > 0.0f ? v : 0.0f;
                p[r] += v * an;
            }
        }
        // Reduce over the 16 lanes of each half (xor masks < 16 stay in-group).
#pragma unroll
        for (int r = 0; r < 8; ++r) {
#pragma unroll
            for (int m = 1; m <= 8; m <<= 1)
                p[r] += __shfl_xor(p[r], m, 32);
        }
        if (l16 == 0) {
#pragma unroll
            for (int r = 0; r < 8; ++r) {
                int node = wave * 16 + half * 8 + r;
                if (node < n_nodes) sout[node] = p[r];
            }
        }
    }
}

// ---------------------------------------------------------------------------
// Kernel 2: zero the atomic accumulation targets (rowsum, out matrix).
// ---------------------------------------------------------------------------
__global__ void init_kernel(float* __restrict__ rowsum, float* __restrict__ outmat,
                            int n, int total_out)
{
    const int stride = gridDim.x * blockDim.x;
    for (int i = blockIdx.x * blockDim.x + threadIdx.x; i < n; i += stride)
        rowsum[i] = 0.0f;
    for (int i = blockIdx.x * blockDim.x + threadIdx.x; i < total_out; i += stride)
        outmat[i] = 0.0f;
}

// ---------------------------------------------------------------------------
// Kernel 3: per-edge attention score -> exp -> atomic rowsum.
// ---------------------------------------------------------------------------
__global__ void edge_att_kernel(const int* __restrict__ row, const int* __restrict__ col,
                                const float* __restrict__ s1, const float* __restrict__ s2,
                                const float* __restrict__ attb,
                                float* __restrict__ exp_att, float* __restrict__ rowsum,
                                int E)
{
    const float ab = attb[0];
    const int stride = gridDim.x * blockDim.x;
    for (int e = blockIdx.x * blockDim.x + threadIdx.x; e < E; e += stride) {
        const int r = row[e];
        const int c = col[e];
        const float ea = __expf(s1[r] + s2[c] + ab);
        exp_att[e] = ea;
        __hip_atomic_fetch_add(&rowsum[r], ea, __ATOMIC_RELAXED, __HIP_MEMORY_SCOPE_AGENT);
    }
}

// ---------------------------------------------------------------------------
// Kernel 4: normalize, blend with adjacency, scatter-SpMM with f32 atomics.
// One wave per edge; 64 hidden dims = 2 atomics per lane (coalesced 128B).
// ---------------------------------------------------------------------------
__global__ void edge_spmm_kernel(const int* __restrict__ row, const int* __restrict__ col,
                                 const float* __restrict__ adj,
                                 const float* __restrict__ embeds,
                                 const float* __restrict__ exp_att,
                                 const float* __restrict__ rowsum,
                                 float* __restrict__ values, float* __restrict__ outmat,
                                 int E)
{
    const int wave   = blockIdx.x * (blockDim.x >> 5) + (threadIdx.x >> 5);
    const int nwaves = gridDim.x * (blockDim.x >> 5);
    const int lane   = threadIdx.x & 31;
    const float inv1pr = 1.0f / (1.0f + RES_F);

    for (int e = wave; e < E; e += nwaves) {
        const int r = row[e];
        const int c = col[e];
        const float v = (exp_att[e] / (rowsum[r] + 1e-6f) + RES_F * adj[e]) * inv1pr;
        if (lane == 0) values[e] = v;
        const float x0 = embeds[(size_t)c * HIDDEN + lane];
        const float x1 = embeds[(size_t)c * HIDDEN + 32 + lane];
        __hip_atomic_fetch_add(&outmat[(size_t)r * HIDDEN + lane], v * x0,
                               __ATOMIC_RELAXED, __HIP_MEMORY_SCOPE_AGENT);
        __hip_atomic_fetch_add(&outmat[(size_t)r * HIDDEN + 32 + lane], v * x1,
                               __ATOMIC_RELAXED, __HIP_MEMORY_SCOPE_AGENT);
    }
}

// ---------------------------------------------------------------------------
extern "C" void kernel_launch(void* const* d_in, const int* in_sizes, int n_in,
                              void* d_out, int out_size, void* d_ws, size_t ws_size,
                              hipStream_t stream)
{
    const int*   edge_index = (const int*)  d_in[0];   // [2, E]
    const float* adj        = (const float*)d_in[1];   // [E]
    const float* embeds     = (const float*)d_in[2];   // [N, 64]
    const float* W1         = (const float*)d_in[3];
    const float* b1         = (const float*)d_in[4];
    const float* W2         = (const float*)d_in[5];
    const float* b2         = (const float*)d_in[6];
    const float* attw       = (const float*)d_in[7];   // [128]
    const float* attb       = (const float*)d_in[8];   // [1]

    const int E = in_sizes[1];
    const int N = in_sizes[2] / HIDDEN;

    const int* row = edge_index;
    const int* col = edge_index + E;

    // Workspace layout (f32): s1[N] | s2[N] | rowsum[N] | exp_att[E]  (~3.8 MB)
    float* s1      = (float*)d_ws;
    float* s2      = s1 + N;
    float* rowsum  = s2 + N;
    float* exp_att = rowsum + N;

    // Output layout: values[E] | out[N*64]
    float* values = (float*)d_out;
    float* outmat = values + E;

    // 1) node transform (WMMA): 1 wave per 16-node tile, 4 waves per block
    const int n_tiles = (N + 15) / 16;
    const int nb_blocks = (n_tiles + 3) / 4;
    node_wmma_kernel<<<nb_blocks, 128, 0, stream>>>(embeds, W1, b1, W2, b2, attw,
                                                    s1, s2, N, n_tiles);

    // 2) zero atomic targets
    init_kernel<<<2048, 256, 0, stream>>>(rowsum, outmat, N, N * HIDDEN);

    // 3) per-edge exp(att) + rowsum
    int att_blocks = (E + 255) / 256;
    if (att_blocks > 4096) att_blocks = 4096;
    edge_att_kernel<<<att_blocks, 256, 0, stream>>>(row, col, s1, s2, attb,
                                                    exp_att, rowsum, E);

    // 4) normalize + scatter SpMM (1 wave per edge, grid-stride)
    int spmm_blocks = (E + 7) / 8;
    if (spmm_blocks > 16384) spmm_blocks = 16384;
    edge_spmm_kernel<<<spmm_blocks, 256, 0, stream>>>(row, col, adj, embeds,
                                                      exp_att, rowsum,
                                                      values, outmat, E);
}